// PotentialNeuralNet_12652973654183
// MI455X (gfx1250) — compile-verified
//
#include <hip/hip_runtime.h>

#define NTA 65536
#define NC  1024
#define NO  256
#define HD  332
#define HP  336   // HD padded to multiple of 16
#define NS  4

#define FS 258   // feat LDS row stride (floats; even for b64, != 0 mod 64)
#define GS 338   // gz   LDS row stride (floats; >= HP, even)

typedef __attribute__((ext_vector_type(2))) float v2f;
typedef __attribute__((ext_vector_type(8))) float v8f;

__device__ __forceinline__ float rsum16(float v) {
  // sum across the 16-lane N-group (wave32; xor masks 1,2,4,8 stay in-group)
  v += __shfl_xor(v, 1, 32);
  v += __shfl_xor(v, 2, 32);
  v += __shfl_xor(v, 4, 32);
  v += __shfl_xor(v, 8, 32);
  return v;
}

__global__ void reann_zero_E(float* __restrict__ E) {
  int i = blockIdx.x * blockDim.x + threadIdx.x;
  if (i < NC) E[i] = 0.f;
}

// Forward weights, k-pair interleaved + zero-padded:
//   W1pF[s][p][n] = { W1[s][2p][n], W1[s][2p+1][n] }   (p = 0..NO/2-1, n = 0..HP-1)
__global__ void reann_prep_w1f(const float* __restrict__ W1, v2f* __restrict__ W1pF) {
  int idx = blockIdx.x * blockDim.x + threadIdx.x;   // over S*(NO/2)*HP
  if (idx >= NS * (NO / 2) * HP) return;
  int n = idx % HP;
  int p = (idx / HP) % (NO / 2);
  int s = idx / (HP * (NO / 2));
  v2f v;
  v.x = (n < HD) ? W1[((size_t)s * NO + 2 * p)     * HD + n] : 0.f;
  v.y = (n < HD) ? W1[((size_t)s * NO + 2 * p + 1) * HD + n] : 0.f;
  W1pF[idx] = v;
}

// Backward (transposed) weights, k-pair interleaved + zero-padded:
//   W1TpF[s][q][n] = { W1^T[2q][n], W1^T[2q+1][n] } = { W1[s][n][2q], W1[s][n][2q+1] }
__global__ void reann_prep_w1t(const float* __restrict__ W1, v2f* __restrict__ W1TpF) {
  int idx = blockIdx.x * blockDim.x + threadIdx.x;   // over S*(HP/2)*NO
  if (idx >= NS * (HP / 2) * NO) return;
  int n = idx % NO;
  int q = (idx / NO) % (HP / 2);
  int s = idx / (NO * (HP / 2));
  int k0 = 2 * q, k1 = 2 * q + 1;
  const float* row = W1 + ((size_t)s * NO + n) * HD;
  v2f v;
  v.x = (k0 < HD) ? row[k0] : 0.f;
  v.y = (k1 < HD) ? row[k1] : 0.f;
  W1TpF[idx] = v;
}

__global__ void reann_prep_vec(const float* __restrict__ b1,
                               const float* __restrict__ W2,
                               float* __restrict__ b1p,
                               float* __restrict__ W2p) {
  int idx = blockIdx.x * blockDim.x + threadIdx.x;   // over S*HP
  if (idx >= NS * HP) return;
  int n = idx % HP;
  int s = idx / HP;
  b1p[idx] = (n < HD) ? b1[s * HD + n] : 0.f;
  W2p[idx] = (n < HD) ? W2[s * HD + n] : 0.f;
}

__launch_bounds__(32)
__global__ void reann_fused(const int*   __restrict__ symbols,
                            const float* __restrict__ positions,
                            const int*   __restrict__ crystalidx,
                            const float* __restrict__ Wd,      // [3][NO]
                            const v2f*   __restrict__ W1pF,    // [S][NO/2][HP]
                            const v2f*   __restrict__ W1TpF,   // [S][HP/2][NO]
                            const float* __restrict__ b1p,     // [S][HP]
                            const float* __restrict__ W2p,     // [S][HP]
                            const float* __restrict__ b2,      // [S]
                            float* __restrict__ en_out,        // [NTA]
                            float* __restrict__ E_out,         // [NC]
                            float* __restrict__ f_out) {       // [NTA][3]
  __shared__ float feat[16 * FS];
  __shared__ float gz[16 * GS];
  __shared__ int   sym[16];

  const int  lane   = threadIdx.x;      // 0..31
  const int  atom0  = blockIdx.x * 16;
  const int  lo     = lane & 15;
  const bool hiHalf = (lane >= 16);
  const int  koff   = hiHalf ? 2 : 0;   // K offset of this lane-half in A/B frags
  const int  kh     = hiHalf ? 1 : 0;   // pair offset (koff/2)

  // ---- stage 1: feat[a][j] = tanh(P[a] . Wd[:,j]) ----
  if (lane < 16) sym[lane] = symbols[atom0 + lane];
  for (int idx = lane; idx < 16 * NO; idx += 32) {
    int a = idx >> 8;           // NO == 256
    int j = idx & (NO - 1);
    const float* p = positions + (size_t)(atom0 + a) * 3;
    float u = p[0] * Wd[j] + p[1] * Wd[NO + j] + p[2] * Wd[2 * NO + j];
    feat[a * FS + j] = tanhf(u);
  }
  __syncthreads();

  int symRow[8];
#pragma unroll
  for (int r = 0; r < 8; ++r) symRow[r] = sym[hiHalf ? (8 + r) : r];

  float fAcc[8][3];
#pragma unroll
  for (int r = 0; r < 8; ++r) { fAcc[r][0] = 0.f; fAcc[r][1] = 0.f; fAcc[r][2] = 0.f; }

  for (int s = 0; s < NS; ++s) {
    // uniform (SGPR) per-species bases; 32-bit lane indices -> GVS scale_offset loads
    const v2f* W1s  = W1pF  + (size_t)s * (NO / 2) * HP;
    const v2f* W1Ts = W1TpF + (size_t)s * (HP / 2) * NO;
    float esAcc[8];
#pragma unroll
    for (int r = 0; r < 8; ++r) esAcc[r] = 0.f;

    // ---------- forward: z = feat @ W1[s]  (M=16, N=HP -> 21 tiles, K=NO) ----------
    for (int t = 0; t < HP / 16; ++t) {
      const int n = t * 16 + lo;
      v8f acc = {0.f, 0.f, 0.f, 0.f, 0.f, 0.f, 0.f, 0.f};
      int bidx = kh * HP + n;            // pair index: (2c+kh)*HP + n, c = 0
#pragma unroll 4
      for (int c = 0; c < NO / 4; ++c) {
        const int k0 = c * 4 + koff;
        v2f a;
        a.x = feat[lo * FS + k0];
        a.y = feat[lo * FS + k0 + 1];
        v2f b = W1s[bidx];               // {W1[k0][n], W1[k0+1][n]}
        bidx += 2 * HP;
        acc = __builtin_amdgcn_wmma_f32_16x16x4_f32(false, a, false, b,
                                                    (short)0, acc, false, false);
      }
      const float b1v = b1p[s * HP + n];
      const float w2v = W2p[s * HP + n];
#pragma unroll
      for (int r = 0; r < 8; ++r) {
        const int row = hiHalf ? (8 + r) : r;      // C layout: VGPR r -> M=r / r+8
        float zb  = acc[r] + b1v;
        float sig = 1.f / (1.f + expf(-zb));
        esAcc[r] += zb * sig * w2v;                               // silu(z) * W2
        gz[row * GS + n] = w2v * sig * (1.f + zb * (1.f - sig));  // W2 * silu'(z)
      }
    }

    // per-atom energy: reduce over N-group lanes, select by species, emit
#pragma unroll
    for (int r = 0; r < 8; ++r) {
      float v = rsum16(esAcc[r]);
      if (lo == r) {
        int atom = hiHalf ? (8 + r) : r;
        if (symRow[r] == s) {
          float e = v + b2[s];
          en_out[atom0 + atom] = e;
          atomicAdd(&E_out[crystalidx[atom0 + atom]], e);
        }
      }
    }
    __syncthreads();  // gz visible to all lanes

    // ---------- backward: gfeat = gz @ W1[s]^T  (N=NO -> 16 tiles, K=HP=84*4) ----------
    for (int o = 0; o < NO / 16; ++o) {
      const int n = o * 16 + lo;  // output feature index (< 256 always)
      v8f acc = {0.f, 0.f, 0.f, 0.f, 0.f, 0.f, 0.f, 0.f};
      int bidx = kh * NO + n;            // pair index: (2c+kh)*NO + n, c = 0
#pragma unroll 4
      for (int c = 0; c < HP / 4; ++c) {
        const int k0 = c * 4 + koff;
        v2f a;
        a.x = gz[lo * GS + k0];
        a.y = gz[lo * GS + k0 + 1];
        v2f b = W1Ts[bidx];              // {W1T[k0][n], W1T[k0+1][n]}
        bidx += 2 * NO;
        acc = __builtin_amdgcn_wmma_f32_16x16x4_f32(false, a, false, b,
                                                    (short)0, acc, false, false);
      }
      const float wd0 = Wd[n], wd1 = Wd[NO + n], wd2 = Wd[2 * NO + n];
#pragma unroll
      for (int r = 0; r < 8; ++r) {
        const int row = hiHalf ? (8 + r) : r;
        float msk = (symRow[r] == s) ? 1.f : 0.f;
        float f = feat[row * FS + n];
        float g = acc[r] * (1.f - f * f) * msk;    // dtanh chain + species select
        fAcc[r][0] += g * wd0;
        fAcc[r][1] += g * wd1;
        fAcc[r][2] += g * wd2;
      }
    }
    __syncthreads();  // finish reading gz before next species overwrites it
  }

  // ---- forces: reduce across N-group lanes and write ----
#pragma unroll
  for (int r = 0; r < 8; ++r) {
    float f0 = rsum16(fAcc[r][0]);
    float f1 = rsum16(fAcc[r][1]);
    float f2 = rsum16(fAcc[r][2]);
    if (lo == r) {
      int atom = hiHalf ? (8 + r) : r;
      float* fp = f_out + (size_t)(atom0 + atom) * 3;
      fp[0] = f0; fp[1] = f1; fp[2] = f2;
    }
  }
}

extern "C" void kernel_launch(void* const* d_in, const int* in_sizes, int n_in,
                              void* d_out, int out_size, void* d_ws, size_t ws_size,
                              hipStream_t stream) {
  // setup_inputs() order:
  // 0 symbols(i32) 1 positions(f32) 2 cells 3 pbcs 4 energyidx 5 crystalidx(i32)
  // 6 Wd 7 W1 8 b1 9 W2 10 b2
  const int*   symbols    = (const int*)d_in[0];
  const float* positions  = (const float*)d_in[1];
  const int*   crystalidx = (const int*)d_in[5];
  const float* Wd = (const float*)d_in[6];
  const float* W1 = (const float*)d_in[7];
  const float* b1 = (const float*)d_in[8];
  const float* W2 = (const float*)d_in[9];
  const float* b2 = (const float*)d_in[10];

  float* out    = (float*)d_out;
  float* en     = out;               // [NTA]
  float* E      = out + NTA;         // [NC]
  float* forces = out + NTA + NC;    // [NTA*3]

  // workspace carve-out (~2.76 MB): pair-interleaved padded weights
  v2f*   W1pF  = (v2f*)d_ws;                               // S*(NO/2)*HP pairs
  v2f*   W1TpF = W1pF + (size_t)NS * (NO / 2) * HP;        // S*(HP/2)*NO pairs
  float* b1p   = (float*)(W1TpF + (size_t)NS * (HP / 2) * NO);  // S*HP
  float* W2p   = b1p + (size_t)NS * HP;                    // S*HP

  reann_prep_w1f<<<(NS * (NO / 2) * HP + 255) / 256, 256, 0, stream>>>(W1, W1pF);
  reann_prep_w1t<<<(NS * (HP / 2) * NO + 255) / 256, 256, 0, stream>>>(W1, W1TpF);
  reann_prep_vec<<<(NS * HP + 255) / 256, 256, 0, stream>>>(b1, W2, b1p, W2p);
  reann_zero_E  <<<(NC + 255) / 256, 256, 0, stream>>>(E);

  reann_fused<<<NTA / 16, 32, 0, stream>>>(symbols, positions, crystalidx,
                                           Wd, W1pF, W1TpF, b1p, W2p, b2,
                                           en, E, forces);
}